// TtMoeLayer_17403207483731
// MI455X (gfx1250) — compile-verified
//
#include <hip/hip_runtime.h>
#include <math.h>

// MoE layer: S=1, B=32, H=4096, E=8, fp32.
// out[b,:] = sum_e wts[b,e] * (x[b,:] @ W_e)  ==  (wts-scaled, expert-stacked x) @ Wstack
//   A[32 x 32768], Wstack[32768 x 4096] (= expert_w viewed flat, row-major)
// Memory-bound: 512MB weight stream / 23.3 TB/s ~ 22us floor.

typedef __attribute__((ext_vector_type(2))) float v2f;
typedef __attribute__((ext_vector_type(8))) float v8f;

#define HD   4096
#define NE   8
#define NB   32
#define KTOT (NE * HD)          // 32768
#define KQ   (KTOT / 4)         // 8192 K-quads
#define NEG_MAX (-3.402823466e+38f)

// ---------------- Kernel 1: gating (logits -> top-2 weights) ----------------
__global__ __launch_bounds__(256) void moe_gate(const float* __restrict__ x,
                                                const float* __restrict__ gate_w,
                                                float* __restrict__ wts) {
    __shared__ float lg[NB][NE];
    const int t = threadIdx.x;
    const int b = t >> 3;
    const int e = t & 7;
    float acc = 0.0f;
    for (int h = 0; h < HD; ++h)
        acc += x[b * HD + h] * gate_w[h * NE + e];
    lg[b][e] = acc;
    __syncthreads();
    if (t < NB) {
        float l[NE];
        for (int i = 0; i < NE; ++i) l[i] = lg[t][i];
        float w0 = l[0];
        for (int i = 1; i < NE; ++i) w0 = fmaxf(w0, l[i]);
        float w1 = NEG_MAX;                         // max over entries masked at ==w0
        for (int i = 0; i < NE; ++i) {
            float v = (l[i] == w0) ? NEG_MAX : l[i];
            w1 = fmaxf(w1, v);
        }
        float pre = 1.0f / (1.0f + __expf(w1 - w0)); // softmax weight of top-1
        for (int i = 0; i < NE; ++i) {
            float wv = (l[i] == w0) ? pre : ((l[i] == w1) ? (1.0f - pre) : 0.0f);
            wts[t * NE + i] = wv;
        }
    }
}

// ---- Kernel 2: build A = wts-scaled x, pre-swizzled into WMMA A-fragment order ----
// Fragment layout (per ISA 16x4 f32 A): element g encodes
//   j  = g & 1            (VGPR 0/1  -> K lsb)
//   L  = (g >> 1) & 31    (lane)
//   mt = (g >> 6) & 1     (M-tile: rows 0-15 / 16-31)
//   kq = g >> 7           (K-quad)
// with  k = kq*4 + 2*(L>>4) + j,   b = mt*16 + (L&15)
__global__ __launch_bounds__(256) void moe_pack_a(const float* __restrict__ x,
                                                  const float* __restrict__ wts,
                                                  float* __restrict__ afrag) {
    const unsigned g  = blockIdx.x * 256u + threadIdx.x;   // < 1<<20
    const unsigned j  = g & 1u;
    const unsigned L  = (g >> 1) & 31u;
    const unsigned mt = (g >> 6) & 1u;
    const unsigned kq = g >> 7;
    const unsigned k  = kq * 4u + 2u * (L >> 4) + j;
    const unsigned b  = mt * 16u + (L & 15u);
    const unsigned e  = k >> 12;       // k / 4096
    const unsigned h  = k & 4095u;     // k % 4096
    afrag[g] = wts[b * NE + e] * x[b * HD + h];
}

// ---------------- Kernel 3: A[32 x 32768] @ Wstack[32768 x 4096] ----------------
// One wave32 per workgroup; wave owns 16 output columns, both 16-row M-tiles.
__global__ __launch_bounds__(32) void moe_gemm(const float* __restrict__ afrag,
                                               const float* __restrict__ wstack,
                                               float* __restrict__ out) {
    const int lane = threadIdx.x;              // wave32
    const int n0   = blockIdx.x * 16;
    const int col  = n0 + (lane & 15);
    const int krow = 2 * (lane >> 4);          // half-wave K split within the quad

    v8f c0 = {};                               // rows 0..15
    v8f c1 = {};                               // rows 16..31

    const float* wp = wstack + (size_t)krow * HD + col;
    const float* ap = afrag + lane * 2;

    for (int kq = 0; kq < KQ; ++kq) {
        // A fragments: one b64 per lane each (pre-swizzled layout)
        v2f a0 = *(const v2f*)(ap);            // M-tile 0
        v2f a1 = *(const v2f*)(ap + 64);       // M-tile 1
        // B fragment: rows k, k+1 of the 16-column strip (coalesced 64B per half-wave)
        v2f bfr;
        bfr.x = wp[0];
        bfr.y = wp[HD];
        // Stream-prefetch the weight tile ~16 K-quads (1 MB/WG window) ahead, clamped.
        const float* pfp = (kq + 16 < KQ) ? (wp + 16 * 4 * HD) : wp;
        __builtin_prefetch(pfp, 0, 1);

        c0 = __builtin_amdgcn_wmma_f32_16x16x4_f32(false, a0, false, bfr,
                                                   (short)0, c0, false, false);
        c1 = __builtin_amdgcn_wmma_f32_16x16x4_f32(false, a1, false, bfr,
                                                   (short)0, c1, false, false);
        ap += 128;            // next K-quad of A fragments
        wp += 4 * HD;         // next 4 rows of Wstack
    }

    // C/D layout: VGPR r -> M = 8*(lane>>4) + r, N = lane&15 (+n0)
    const int rbase = (lane >> 4) * 8;
    for (int r = 0; r < 8; ++r) {
        out[(size_t)(rbase + r) * HD + col]      = c0[r];
        out[(size_t)(16 + rbase + r) * HD + col] = c1[r];
    }
}

extern "C" void kernel_launch(void* const* d_in, const int* in_sizes, int n_in,
                              void* d_out, int out_size, void* d_ws, size_t ws_size,
                              hipStream_t stream) {
    const float* x        = (const float*)d_in[0];   // [1,1,32,4096]
    const float* gate_w   = (const float*)d_in[1];   // [4096,8]
    const float* expert_w = (const float*)d_in[2];   // [8,4096,4096] == Wstack[32768,4096]
    float* out = (float*)d_out;                      // [1,1,32,4096]

    float* wts   = (float*)d_ws;                     // 256 floats
    float* afrag = (float*)((char*)d_ws + 1024);     // 4 MB A-fragment buffer

    moe_gate<<<1, 256, 0, stream>>>(x, gate_w, wts);
    moe_pack_a<<<(NB * KTOT) / 256, 256, 0, stream>>>(x, wts, afrag);
    moe_gemm<<<HD / 16, 32, 0, stream>>>(afrag, expert_w, out);
}